// net_PhyNet_65455301591501
// MI455X (gfx1250) — compile-verified
//
#include <hip/hip_runtime.h>

#define HIS 16
#define PRE 25
#define NNEI 64
#define TDT 0.1f

typedef float v2f __attribute__((ext_vector_type(2)));
typedef float v4f __attribute__((ext_vector_type(4)));
typedef float v8f __attribute__((ext_vector_type(8)));

struct PP {
  const float *emb_W, *emb_b, *fs_W, *fs_b, *fm_W, *fm_b, *de_W, *de_b;
  const float *zs_W, *zs_b, *zm_W, *zm_b; // null for mv nets
};
struct Args {
  const float* history;  // [B,16,4]
  const float* frames;   // [B,25,64,3]
  float* out;            // [B,25,4]
  PP mv, mvv, sf, sfv;
};

__global__ __launch_bounds__(128) void phynet_kernel(Args args)
{
  const int tid     = threadIdx.x;
  const int lane    = tid & 31;
  const int wave    = tid >> 5;     // wave = M-tile (rows 16w..16w+15)
  const int lane15  = lane & 15;
  const int halfSel = lane >> 4;    // 0: K=0,1 half; 1: K=2,3 half
  const int agent   = blockIdx.x;

  // ---------------- LDS (~26 KB) ----------------
  __shared__ v2f  sBF[2][2][128];      // sf B-fragment image [q][half][col] (static)
  __shared__ v2f  sAFrag[128];         // sf A-fragment image [half*64+i]
  __shared__ v2f  sPQf[2][2][64];      // mv B-fragment image [q][half][col]
  __shared__ v4f  sPackS[2][64];       // sf mf:   {bias(step), w_sin, w_cos, 0}
  __shared__ v4f  sPackZ[2][64];       // sf zero: {zs_b, zm_sin, zm_cos, 0} (static)
  __shared__ v4f  sPackM[2][64];       // mv:      {fs_b, fm_sin, fm_cos, 0} (static)
  __shared__ float sB0[2][64], sU0[2][64], sU1[2][64];
  __shared__ float sMvFs[2][16][64];
  __shared__ float sMvDeW[2][128];
  __shared__ float sFmB[2], sZmB[2], sDeS[2][2], sDeB[2][2];
  __shared__ float sMvFmB[2], sMvDeB[2][2];
  __shared__ float sTraj[HIS + PRE][4];
  __shared__ float sMf[2][64], sZr[2][64], sMfMv[2][64];
  __shared__ float sRes[8];            // mv,mvv,sf,sfv (x,y)

  // ---------------- one-time weight folding ----------------
  for (int q = 0; q < 2; ++q) {
    const PP p = q ? args.sfv : args.sf;
    {   // folded rank-3 rows -> B-fragment image
      const int n = tid;   // 0..127
      float a0 = 0.f, a1 = 0.f, a2 = 0.f;
      for (int j = 0; j < 64; ++j) {
        float wc = (n < 64) ? p.fs_W[(64 + j) * 64 + n] : p.zs_W[j * 64 + (n - 64)];
        a0 += p.emb_W[j]      * wc;
        a1 += p.emb_W[64 + j] * wc;
        a2 += p.emb_b[j]      * wc;
      }
      v2f h0 = { a0, a1 };  v2f h1 = { a2, 0.f };
      sBF[q][0][n] = h0;    sBF[q][1][n] = h1;
    }
    if (tid < 64) { // pos-branch fold over fs_top + static packs
      const int j = tid;
      float u0 = 0.f, u1 = 0.f, ub = 0.f;
      for (int k = 0; k < 64; ++k) {
        float w = p.fs_W[k * 64 + j];
        u0 += p.emb_W[k] * w; u1 += p.emb_W[64 + k] * w; ub += p.emb_b[k] * w;
      }
      sU0[q][j] = u0; sU1[q][j] = u1; sB0[q][j] = p.fs_b[j] + ub;
      v4f pz = { p.zs_b[j], p.zm_W[j], p.zm_W[64 + j], 0.f };
      sPackZ[q][j] = pz;
      v4f ps = { 0.f, p.fm_W[j], p.fm_W[64 + j], 0.f };   // .x refreshed per step
      sPackS[q][j] = ps;
    }
    if (tid == 0) {
      sFmB[q] = p.fm_b[0]; sZmB[q] = p.zm_b[0];
      float s0 = 0.f, s1 = 0.f;
      for (int e = 0; e < 64; ++e) { s0 += p.de_W[2 * e]; s1 += p.de_W[2 * e + 1]; }
      sDeS[q][0] = s0; sDeS[q][1] = s1;
      sDeB[q][0] = p.de_b[0]; sDeB[q][1] = p.de_b[1];
    }
  }
  for (int q = 0; q < 2; ++q) {
    const PP p = q ? args.mvv : args.mv;
    for (int x = tid; x < 16 * 64; x += 128) (&sMvFs[q][0][0])[x] = p.fs_W[x];
    if (tid < 64) {
      const int j = tid;
      float s = 0.f;
      for (int k = 0; k < 16; ++k) s += p.fs_W[k * 64 + j];
      v2f h1 = { s, 0.f };
      sPQf[q][1][j] = h1;                       // static half of mv B fragments
      v4f pm = { p.fs_b[j], p.fm_W[j], p.fm_W[64 + j], 0.f };
      sPackM[q][j] = pm;
    }
    sMvDeW[q][tid] = p.de_W[tid];
    if (tid == 0) {
      sMvFmB[q] = p.fm_b[0];
      sMvDeB[q][0] = p.de_b[0]; sMvDeB[q][1] = p.de_b[1];
    }
  }
  if (tid < 64) {                               // static half of sf A fragments
    v2f one = { 1.f, 0.f };
    sAFrag[64 + tid] = one;
  }
  // trajectory ring: rows 0..15 = history, 16..40 = predictions (init 0)
  if (tid < 64) (&sTraj[0][0])[tid] = args.history[(size_t)agent * 64 + tid];
  for (int x = tid; x < PRE * 4; x += 128) (&sTraj[0][0])[64 + x] = 0.f;

  // mv A fragments: rows e -> [embW0[e], embW1[e], emb_b[e], 0] (register select,
  // loads unconditional so no exec-masked memory ops)
  float mvAx[2], mvAy[2];
  {
    const int e = 16 * wave + lane15;
    const PP* mp[2] = { &args.mv, &args.mvv };
    for (int q = 0; q < 2; ++q) {
      float w0 = mp[q]->emb_W[e], w1 = mp[q]->emb_W[64 + e], bb = mp[q]->emb_b[e];
      mvAx[q] = halfSel ? bb  : w0;
      mvAy[q] = halfSel ? 0.f : w1;
    }
  }
  __syncthreads();

  // ---------------- sequential 25-step scan ----------------
  for (int t = 0; t < PRE; ++t) {
    const int start = (t < HIS) ? t : PRE;      // reference's window rule

    // ---- phase 1: frame load, mv P/Q, sf per-step bias ----
    if (tid < NNEI) {
      const float* f = &args.frames[(((size_t)agent * PRE + t) * NNEI + tid) * 3];
      v2f fr = { f[0], f[1] };
      sAFrag[tid] = fr;
      if (t + 1 < PRE)
        __builtin_prefetch(&args.frames[(((size_t)agent * PRE + t + 1) * NNEI + tid) * 3], 0, 0);
    }
    {
      const int q = tid >> 6, j = tid & 63;
      float pa = 0.f, qa = 0.f;
      for (int k = 0; k < 16; ++k) {
        float hx = sTraj[start + k][0], hy = sTraj[start + k][1];
        float w  = sMvFs[q][k][j];
        pa += hx * w; qa += hy * w;
      }
      v2f pq = { pa, qa };
      sPQf[q][0][j] = pq;
      const float lpx = sTraj[start + 15][0], lpy = sTraj[start + 15][1];
      sPackS[q][j].x = sB0[q][j] + lpx * sU0[q][j] + lpy * sU1[q][j];
    }
    __syncthreads();

    // ---- phase 2: fragment work for all four nets (no barriers inside) ----
    // social-force nets: WMMA rank-3 expansion + in-fragment Fourier map
    for (int q = 0; q < 2; ++q) {
      v2f Af = sAFrag[halfSel * 64 + 16 * wave + lane15];
      float accMf[8] = {0}, accZr[8] = {0};
      #pragma unroll
      for (int n = 0; n < 8; ++n) {              // n<4 -> mf tiles, n>=4 -> zero tiles
        const int col = 16 * n + lane15;
        v2f Bf = sBF[q][halfSel][col];
        v8f C = {};
        C = __builtin_amdgcn_wmma_f32_16x16x4_f32(false, Af, false, Bf,
                                                  (short)0, C, false, false);
        v4f pk = (n < 4) ? sPackS[q][col] : sPackZ[q][col - 64];
        float* acc = (n < 4) ? accMf : accZr;
        #pragma unroll
        for (int v = 0; v < 8; ++v) {
          float h = C[v] + pk.x;
          acc[v] += __sinf(h) * pk.y + __cosf(h) * pk.z;
        }
      }
      #pragma unroll
      for (int off = 1; off < 16; off <<= 1) {   // reduce 16-lane column groups
        #pragma unroll
        for (int v = 0; v < 8; ++v) {
          accMf[v] += __shfl_xor(accMf[v], off, 32);
          accZr[v] += __shfl_xor(accZr[v], off, 32);
        }
      }
      if (lane15 == 0) {
        const int r = 16 * wave + 8 * halfSel;
        #pragma unroll
        for (int v = 0; v < 8; ++v) {
          sMf[q][r + v] = accMf[v] + sFmB[q];
          sZr[q][r + v] = accZr[v] + sZmB[q];
        }
      }
    }
    // motivation nets: same rank-3 WMMA trick, K rows [P_t; Q_t; S; 0]
    for (int q = 0; q < 2; ++q) {
      v2f Af = { mvAx[q], mvAy[q] };
      float acc[8] = {0};
      #pragma unroll
      for (int n = 0; n < 4; ++n) {
        const int col = 16 * n + lane15;
        v2f Bf = sPQf[q][halfSel][col];
        v8f C = {};
        C = __builtin_amdgcn_wmma_f32_16x16x4_f32(false, Af, false, Bf,
                                                  (short)0, C, false, false);
        v4f pk = sPackM[q][col];
        #pragma unroll
        for (int v = 0; v < 8; ++v) {
          float h = C[v] + pk.x;
          acc[v] += __sinf(h) * pk.y + __cosf(h) * pk.z;
        }
      }
      #pragma unroll
      for (int off = 1; off < 16; off <<= 1) {
        #pragma unroll
        for (int v = 0; v < 8; ++v) acc[v] += __shfl_xor(acc[v], off, 32);
      }
      if (lane15 == 0) {
        const int r = 16 * wave + 8 * halfSel;
        #pragma unroll
        for (int v = 0; v < 8; ++v) sMfMv[q][r + v] = acc[v] + sMvFmB[q];
      }
    }
    __syncthreads();

    // ---- phase 3: wave-parallel reductions ----
    if (wave < 2) {                   // sf / sfv totals
      const int q = wave;
      float s1 = sMf[q][lane] + sMf[q][lane + 32];
      float s2 = fmaxf(sZr[q][lane], 0.f) + fmaxf(sZr[q][lane + 32], 0.f);
      #pragma unroll
      for (int off = 1; off < 32; off <<= 1) {
        s1 += __shfl_xor(s1, off, 32);
        s2 += __shfl_xor(s2, off, 32);
      }
      if (lane == 0) {
        sRes[4 + 2 * q + 0] = s1 * sDeS[q][0] + 64.f * sDeB[q][0] + s2;
        sRes[4 + 2 * q + 1] = s1 * sDeS[q][1] + 64.f * sDeB[q][1] + s2;
      }
    } else {                          // mv / mvv de-projection
      const int q = wave - 2;
      float o0 = sMfMv[q][lane] * sMvDeW[q][2 * lane]
               + sMfMv[q][lane + 32] * sMvDeW[q][2 * (lane + 32)];
      float o1 = sMfMv[q][lane] * sMvDeW[q][2 * lane + 1]
               + sMfMv[q][lane + 32] * sMvDeW[q][2 * (lane + 32) + 1];
      #pragma unroll
      for (int off = 1; off < 32; off <<= 1) {
        o0 += __shfl_xor(o0, off, 32);
        o1 += __shfl_xor(o1, off, 32);
      }
      if (lane == 0) {
        sRes[2 * q + 0] = o0 + sMvDeB[q][0];
        sRes[2 * q + 1] = o1 + sMvDeB[q][1];
      }
    }
    __syncthreads();

    // ---- phase 4: integrate ----
    if (tid == 0) {
      const float lpx = sTraj[start + 15][0], lpy = sTraj[start + 15][1];
      const float lvx = sTraj[start + 15][2], lvy = sTraj[start + 15][3];
      const float axc = sRes[0] + sRes[4], ayc = sRes[1] + sRes[5];
      const float avx = sRes[2] + sRes[6], avy = sRes[3] + sRes[7];
      const float px = axc * 0.5f * TDT * TDT + lvx * TDT + lpx;
      const float py = ayc * 0.5f * TDT * TDT + lvy * TDT + lpy;
      const float vx = TDT * avx + lvx;
      const float vy = TDT * avy + lvy;
      sTraj[HIS + t][0] = px; sTraj[HIS + t][1] = py;
      sTraj[HIS + t][2] = vx; sTraj[HIS + t][3] = vy;
      float* o = &args.out[((size_t)agent * PRE + t) * 4];
      o[0] = px; o[1] = py; o[2] = vx; o[3] = vy;
    }
    __syncthreads();
  }
}

extern "C" void kernel_launch(void* const* d_in, const int* in_sizes, int n_in,
                              void* d_out, int out_size, void* d_ws, size_t ws_size,
                              hipStream_t stream)
{
  (void)n_in; (void)out_size; (void)d_ws; (void)ws_size;
  Args a{};
  a.history = (const float*)d_in[0];
  a.frames  = (const float*)d_in[1];
  a.out     = (float*)d_out;
  auto fill = [&](PP& p, int o, bool hasZ) {
    p.emb_W = (const float*)d_in[o + 0]; p.emb_b = (const float*)d_in[o + 1];
    p.fs_W  = (const float*)d_in[o + 2]; p.fs_b  = (const float*)d_in[o + 3];
    p.fm_W  = (const float*)d_in[o + 4]; p.fm_b  = (const float*)d_in[o + 5];
    p.de_W  = (const float*)d_in[o + 6]; p.de_b  = (const float*)d_in[o + 7];
    if (hasZ) {
      p.zs_W = (const float*)d_in[o + 8];  p.zs_b = (const float*)d_in[o + 9];
      p.zm_W = (const float*)d_in[o + 10]; p.zm_b = (const float*)d_in[o + 11];
    }
  };
  fill(a.mv, 2, false);    // mv:  inputs 2..9
  fill(a.mvv, 10, false);  // mvv: inputs 10..17
  fill(a.sf, 18, true);    // sf:  inputs 18..29
  fill(a.sfv, 30, true);   // sfv: inputs 30..41
  const int B = in_sizes[0] / (HIS * 4);
  hipLaunchKernelGGL(phynet_kernel, dim3(B), dim3(128), 0, stream, a);
}